// GraphTransformerLayer_32091995636257
// MI455X (gfx1250) — compile-verified
//
#include <hip/hip_runtime.h>
#include <math.h>

#define D_   128
#define H_   8
#define DK_  16
#define DFF_ 512
#define N_   50000
#define E_   400000

typedef float v2f __attribute__((ext_vector_type(2)));
typedef float v8f __attribute__((ext_vector_type(8)));

#if __has_builtin(__builtin_amdgcn_wmma_f32_16x16x4_f32)
#define USE_WMMA 1
#else
#define USE_WMMA 0
#endif

// ---------- float <-> orderable uint (for atomic max on signed floats) ----------
__device__ __forceinline__ unsigned f2ord(float f) {
    unsigned u = __float_as_uint(f);
    return (u & 0x80000000u) ? ~u : (u | 0x80000000u);
}
__device__ __forceinline__ float ord2f(unsigned u) {
    return __uint_as_float((u & 0x80000000u) ? (u ^ 0x80000000u) : ~u);
}

// ---------- init m(ord)=0 (== -inf class), z=0, acc=0 ----------
__global__ void init_mza_kernel(unsigned* __restrict__ mOrd, float* __restrict__ z,
                                float* __restrict__ acc) {
    int idx = blockIdx.x * blockDim.x + threadIdx.x;
    if (idx < N_ * D_) acc[idx] = 0.f;
    if (idx < N_ * H_) { z[idx] = 0.f; mOrd[idx] = 0u; }
}

// ---------- generic GEMM: Y[nr x Dout] = X[nr x Din] @ W + bias (opt relu) ----------
// one wave per 16x16 output tile; fp32 WMMA 16x16x4
__global__ __launch_bounds__(32) void wmma_gemm_kernel(
    const float* __restrict__ X, const float* __restrict__ W,
    const float* __restrict__ bias, float* __restrict__ Y,
    int Din, int Dout, int relu)
{
    const int rt   = blockIdx.x;
    const int ctn  = blockIdx.y;
    const int lane = threadIdx.x;
    const int hi   = lane >> 4;      // half: 0/1
    const int l    = lane & 15;
    const int col  = ctn * 16 + l;
#if USE_WMMA
    const float* xrow = X + (size_t)(rt * 16 + l) * Din;
    v8f c = {};
    for (int k = 0; k < Din; k += 4) {
        int ka = k + 2 * hi;
        v2f a; a.x = xrow[ka];                       a.y = xrow[ka + 1];
        v2f b; b.x = W[(size_t)ka * Dout + col];     b.y = W[(size_t)(ka + 1) * Dout + col];
        c = __builtin_amdgcn_wmma_f32_16x16x4_f32(false, a, false, b, (short)0, c, false, false);
    }
#pragma unroll
    for (int r = 0; r < 8; ++r) {
        int orow = rt * 16 + r + 8 * hi;
        float v = c[r] + bias[col];
        if (relu) v = fmaxf(v, 0.f);
        Y[(size_t)orow * Dout + col] = v;
    }
#else
    for (int r = 0; r < 8; ++r) {
        int orow = rt * 16 + r + 8 * hi;
        const float* xr = X + (size_t)orow * Din;
        float accv = bias[col];
        for (int k = 0; k < Din; ++k) accv += xr[k] * W[(size_t)k * Dout + col];
        if (relu) accv = fmaxf(accv, 0.f);
        Y[(size_t)orow * Dout + col] = accv;
    }
#endif
}

// ---------- edge scores + segment max (wave per edge) ----------
__global__ __launch_bounds__(256) void edge_scores_kernel(
    const int* __restrict__ ei, const float* __restrict__ Q, const float* __restrict__ K,
    const float* __restrict__ srcX, const float* __restrict__ tgtX,
    const float* __restrict__ We, const float* __restrict__ be,
    float* __restrict__ scores, unsigned* __restrict__ mOrd)
{
    int wid = (blockIdx.x * blockDim.x + threadIdx.x) >> 5;
    if (wid >= E_) return;
    int lane = threadIdx.x & 31;
    int t = ei[wid];
    int s = ei[E_ + wid];
    const float4 sx = ((const float4*)(srcX + (size_t)s * D_))[lane];
    const float4 tx = ((const float4*)(tgtX + (size_t)t * D_))[lane];
    const float4 qv = ((const float4*)(Q    + (size_t)t * D_))[lane];
    const float4 kv = ((const float4*)(K    + (size_t)s * D_))[lane];
    float p[H_];
#pragma unroll
    for (int h = 0; h < H_; ++h) p[h] = 0.f;
    const float sa[4] = {sx.x, sx.y, sx.z, sx.w};
    const float ta[4] = {tx.x, tx.y, tx.z, tx.w};
#pragma unroll
    for (int i = 0; i < 4; ++i) {
        int d = lane * 4 + i;
        const float* w0 = We + (size_t)d * H_;          // src rows 0..127
        const float* w1 = We + (size_t)(D_ + d) * H_;   // tgt rows 128..255
#pragma unroll
        for (int h = 0; h < H_; ++h) p[h] += sa[i] * w0[h] + ta[i] * w1[h];
    }
    // q.k partial: this lane's 4 dims lie in head (lane>>2)
    float qk = qv.x * kv.x + qv.y * kv.y + qv.z * kv.z + qv.w * kv.w;
    qk += __shfl_xor(qk, 1);
    qk += __shfl_xor(qk, 2);
#pragma unroll
    for (int h = 0; h < H_; ++h)
        for (int off = 16; off >= 1; off >>= 1) p[h] += __shfl_xor(p[h], off);
    if ((lane & 3) == 0) {
        int h = lane >> 2;
        float sc = qk * 0.25f + p[h] + be[h];      // 1/sqrt(16) = 0.25
        scores[(size_t)wid * H_ + h] = sc;
        atomicMax(&mOrd[(size_t)t * H_ + h], f2ord(sc));
    }
}

// ---------- exp(score-m), segment-sum z, accumulate e*v into acc (wave per edge) ----------
__global__ __launch_bounds__(256) void edge_accum_kernel(
    const int* __restrict__ ei, const float* __restrict__ scores,
    const unsigned* __restrict__ mOrd, float* __restrict__ z,
    const float* __restrict__ V, float* __restrict__ acc)
{
    int wid = (blockIdx.x * blockDim.x + threadIdx.x) >> 5;
    if (wid >= E_) return;
    int lane = threadIdx.x & 31;
    int t = ei[wid];
    int s = ei[E_ + wid];
    int h = lane >> 2;
    float mm = ord2f(mOrd[(size_t)t * H_ + h]);
    float eh = __expf(scores[(size_t)wid * H_ + h] - mm);
    if ((lane & 3) == 0) atomicAdd(&z[(size_t)t * H_ + h], eh);
    const float4 vv = ((const float4*)(V + (size_t)s * D_))[lane];
    float* ap = acc + (size_t)t * D_ + lane * 4;
    atomicAdd(ap + 0, eh * vv.x);
    atomicAdd(ap + 1, eh * vv.y);
    atomicAdd(ap + 2, eh * vv.z);
    atomicAdd(ap + 3, eh * vv.w);
}

// ---------- acc /= z (0 if no incoming edges, matching segment-sum-of-empty) ----------
__global__ void finalize_attn_kernel(float* __restrict__ acc, const float* __restrict__ z) {
    int idx = blockIdx.x * blockDim.x + threadIdx.x;
    if (idx >= N_ * D_) return;
    float zv = z[(idx >> 7) * H_ + ((idx & 127) >> 4)];
    acc[idx] = (zv > 0.f) ? acc[idx] / zv : 0.f;
}

// ---------- out = LN(x [+ y]) ; wave per row ----------
__global__ __launch_bounds__(256) void row_addln_kernel(
    const float* __restrict__ X, const float* __restrict__ Yopt,
    const float* __restrict__ g, const float* __restrict__ b, float* __restrict__ out)
{
    int row = (blockIdx.x * blockDim.x + threadIdx.x) >> 5;
    if (row >= N_) return;
    int lane = threadIdx.x & 31;
    float4 v = ((const float4*)(X + (size_t)row * D_))[lane];
    if (Yopt) {
        float4 y = ((const float4*)(Yopt + (size_t)row * D_))[lane];
        v.x += y.x; v.y += y.y; v.z += y.z; v.w += y.w;
    }
    float s = v.x + v.y + v.z + v.w;
    for (int off = 16; off >= 1; off >>= 1) s += __shfl_xor(s, off);
    float mu = s * (1.f / D_);
    float4 d = {v.x - mu, v.y - mu, v.z - mu, v.w - mu};
    float sq = d.x * d.x + d.y * d.y + d.z * d.z + d.w * d.w;
    for (int off = 16; off >= 1; off >>= 1) sq += __shfl_xor(sq, off);
    float rstd = rsqrtf(sq * (1.f / D_) + 1e-5f);
    float4 gv = ((const float4*)g)[lane];
    float4 bv = ((const float4*)b)[lane];
    float4 o = {gv.x * d.x * rstd + bv.x, gv.y * d.y * rstd + bv.y,
                gv.z * d.z * rstd + bv.z, gv.w * d.w * rstd + bv.w};
    ((float4*)(out + (size_t)row * D_))[lane] = o;
}

// ---------- fused FFN + residual + LN per 16-row tile; WMMA from LDS ----------
#define HT_S 132   // 132 % 64 == 4 -> conflict-free lane-per-row access
#define F1_S 516
__global__ __launch_bounds__(128) void ffn_ln_kernel(
    float* __restrict__ h,   // in/out: slice of d_out, N x 128
    const float* __restrict__ W1, const float* __restrict__ b1,
    const float* __restrict__ W2, const float* __restrict__ b2,
    const float* __restrict__ g, const float* __restrict__ bb)
{
    __shared__ float hT[16 * HT_S];
    __shared__ float f1[16 * F1_S];
    const int tid  = threadIdx.x;
    const int row0 = blockIdx.x * 16;
    for (int e = tid; e < 16 * D_; e += 128) {
        int r = e >> 7, cc = e & 127;
        hT[r * HT_S + cc] = h[(size_t)(row0 + r) * D_ + cc];
    }
    __syncthreads();
    const int wave = tid >> 5;
    const int lane = tid & 31;
    const int hi   = lane >> 4;
    const int l    = lane & 15;
    // GEMM1: 128 -> 512, relu; 32 col tiles, 8 per wave
#if USE_WMMA
    for (int ct = 0; ct < 8; ++ct) {
        int col = (wave * 8 + ct) * 16 + l;
        v8f c = {};
        for (int k = 0; k < D_; k += 4) {
            int ka = k + 2 * hi;
            v2f a; a.x = hT[l * HT_S + ka];              a.y = hT[l * HT_S + ka + 1];
            v2f b; b.x = W1[(size_t)ka * DFF_ + col];    b.y = W1[(size_t)(ka + 1) * DFF_ + col];
            c = __builtin_amdgcn_wmma_f32_16x16x4_f32(false, a, false, b, (short)0, c, false, false);
        }
#pragma unroll
        for (int r = 0; r < 8; ++r) {
            int rl = r + 8 * hi;
            f1[rl * F1_S + col] = fmaxf(c[r] + b1[col], 0.f);
        }
    }
#else
    for (int ct = 0; ct < 8; ++ct) {
        int col = (wave * 8 + ct) * 16 + l;
        for (int r = 0; r < 8; ++r) {
            int rl = r + 8 * hi;
            float a = b1[col];
            for (int k = 0; k < D_; ++k) a += hT[rl * HT_S + k] * W1[(size_t)k * DFF_ + col];
            f1[rl * F1_S + col] = fmaxf(a, 0.f);
        }
    }
#endif
    __syncthreads();
    // GEMM2: 512 -> 128 + residual; 8 col tiles, 2 per wave (disjoint cols -> no cross-wave hazard)
#if USE_WMMA
    for (int t2 = 0; t2 < 2; ++t2) {
        int col = (wave + t2 * 4) * 16 + l;
        v8f c = {};
        for (int k = 0; k < DFF_; k += 4) {
            int ka = k + 2 * hi;
            v2f a; a.x = f1[l * F1_S + ka];              a.y = f1[l * F1_S + ka + 1];
            v2f b; b.x = W2[(size_t)ka * D_ + col];      b.y = W2[(size_t)(ka + 1) * D_ + col];
            c = __builtin_amdgcn_wmma_f32_16x16x4_f32(false, a, false, b, (short)0, c, false, false);
        }
#pragma unroll
        for (int r = 0; r < 8; ++r) {
            int rl = r + 8 * hi;
            hT[rl * HT_S + col] = c[r] + b2[col] + hT[rl * HT_S + col];
        }
    }
#else
    for (int t2 = 0; t2 < 2; ++t2) {
        int col = (wave + t2 * 4) * 16 + l;
        for (int r = 0; r < 8; ++r) {
            int rl = r + 8 * hi;
            float a = b2[col];
            for (int k = 0; k < DFF_; ++k) a += f1[rl * F1_S + k] * W2[(size_t)k * D_ + col];
            hT[rl * HT_S + col] = a + hT[rl * HT_S + col];
        }
    }
#endif
    __syncthreads();
    // LayerNorm: 8 threads per row
    int r   = tid >> 3;
    int seg = tid & 7;
    float s = 0.f, sq = 0.f;
    for (int i = 0; i < 16; ++i) {
        float v = hT[r * HT_S + seg * 16 + i];
        s += v; sq += v * v;
    }
    s += __shfl_xor(s, 1);  sq += __shfl_xor(sq, 1);
    s += __shfl_xor(s, 2);  sq += __shfl_xor(sq, 2);
    s += __shfl_xor(s, 4);  sq += __shfl_xor(sq, 4);
    float mu   = s * (1.f / D_);
    float rstd = rsqrtf(sq * (1.f / D_) - mu * mu + 1e-5f);
    for (int i = 0; i < 16; ++i) {
        int cc = seg * 16 + i;
        float v = (hT[r * HT_S + cc] - mu) * rstd;
        h[(size_t)(row0 + r) * D_ + cc] = g[cc] * v + bb[cc];
    }
}

extern "C" void kernel_launch(void* const* d_in, const int* in_sizes, int n_in,
                              void* d_out, int out_size, void* d_ws, size_t ws_size,
                              hipStream_t stream)
{
    const float* x_wave  = (const float*)d_in[0];
    const float* x_trans = (const float*)d_in[1];
    const float* x_targ  = (const float*)d_in[2];

    const float *Wq_trans, *bq_trans, *Wk_wave, *bk_wave, *Wv_wave, *bv_wave;
    const float *Wq_targ, *bq_targ, *Wk_trans, *bk_trans, *Wv_trans, *bv_trans;
    const float *We_wt, *be_wt, *We_tt, *be_tt;
    const float *Wout_trans, *bout_trans, *Wout_targ, *bout_targ;
    const float *gA_wave, *bA_wave, *gA_trans, *bA_trans, *gA_targ, *bA_targ;
    const float *W1_wave, *b1_wave, *W2_wave, *b2_wave;
    const float *W1_trans, *b1_trans, *W2_trans, *b2_trans;
    const float *W1_targ, *b1_targ, *W2_targ, *b2_targ;
    const float *gF_wave, *bF_wave, *gF_trans, *bF_trans, *gF_targ, *bF_targ;
    const int *ei_wt, *ei_tt;

    if (n_in >= 57) {
        auto P = [&](int i) { return (const float*)d_in[i]; };
        if (in_sizes[3] == 2048) {
            // jax-pytree (sorted-key) flattening
            We_tt = P(3);  be_tt = P(4);  We_wt = P(5);  be_wt = P(6);
            W1_targ = P(7);  b1_targ = P(8);  W2_targ = P(9);  b2_targ = P(10);
            W1_trans = P(11); b1_trans = P(12); W2_trans = P(13); b2_trans = P(14);
            W1_wave = P(15); b1_wave = P(16); W2_wave = P(17); b2_wave = P(18);
            gA_targ = P(19); bA_targ = P(20); gA_trans = P(21); bA_trans = P(22);
            gA_wave = P(23); bA_wave = P(24);
            gF_targ = P(25); bF_targ = P(26); gF_trans = P(27); bF_trans = P(28);
            gF_wave = P(29); bF_wave = P(30);
            Wout_targ = P(31); bout_targ = P(32); Wout_trans = P(33); bout_trans = P(34);
            // out.wave P(35)/P(36) unused
            Wk_targ_unused: ;
            const float* Wk_targ = P(37); (void)Wk_targ;
            Wq_targ = P(38);  /*Wv_targ P(39)*/ bq_targ = P(41);
            Wk_trans = P(43); Wq_trans = P(44); Wv_trans = P(45);
            bk_trans = P(46); bq_trans = P(47); bv_trans = P(48);
            Wk_wave = P(49);  Wv_wave = P(51);  bk_wave = P(52); bv_wave = P(54);
        } else {
            // insertion-order flattening
            Wq_wave_unused: ;
            Wk_wave = P(5);  bk_wave = P(6);  Wv_wave = P(7);  bv_wave = P(8);
            Wq_trans = P(9); bq_trans = P(10); Wk_trans = P(11); bk_trans = P(12);
            Wv_trans = P(13); bv_trans = P(14);
            Wq_targ = P(15); bq_targ = P(16);
            We_wt = P(21); be_wt = P(22); We_tt = P(23); be_tt = P(24);
            Wout_trans = P(27); bout_trans = P(28); Wout_targ = P(29); bout_targ = P(30);
            gA_wave = P(31); bA_wave = P(32); gA_trans = P(33); bA_trans = P(34);
            gA_targ = P(35); bA_targ = P(36);
            W1_wave = P(37); b1_wave = P(38); W2_wave = P(39); b2_wave = P(40);
            W1_trans = P(41); b1_trans = P(42); W2_trans = P(43); b2_trans = P(44);
            W1_targ = P(45); b1_targ = P(46); W2_targ = P(47); b2_targ = P(48);
            gF_wave = P(49); bF_wave = P(50); gF_trans = P(51); bF_trans = P(52);
            gF_targ = P(53); bF_targ = P(54);
        }
        ei_wt = (const int*)d_in[55];
        ei_tt = (const int*)d_in[56];
    } else {
        // params as single flat blob at d_in[3], insertion order
        const float* blob = (const float*)d_in[3];
        size_t o = 0;
        auto nxt = [&](size_t n) { const float* p = blob + o; o += n; return p; };
        nxt(16384); nxt(128);                       // wave Wq,bq (unused)
        Wk_wave = nxt(16384); bk_wave = nxt(128);
        Wv_wave = nxt(16384); bv_wave = nxt(128);
        Wq_trans = nxt(16384); bq_trans = nxt(128);
        Wk_trans = nxt(16384); bk_trans = nxt(128);
        Wv_trans = nxt(16384); bv_trans = nxt(128);
        Wq_targ = nxt(16384); bq_targ = nxt(128);
        nxt(16384); nxt(128); nxt(16384); nxt(128); // target Wk,bk,Wv,bv (unused)
        We_wt = nxt(2048); be_wt = nxt(8); We_tt = nxt(2048); be_tt = nxt(8);
        nxt(16384); nxt(128);                       // out.wave (unused)
        Wout_trans = nxt(16384); bout_trans = nxt(128);
        Wout_targ = nxt(16384); bout_targ = nxt(128);
        gA_wave = nxt(128); bA_wave = nxt(128);
        gA_trans = nxt(128); bA_trans = nxt(128);
        gA_targ = nxt(128); bA_targ = nxt(128);
        W1_wave = nxt(65536); b1_wave = nxt(512); W2_wave = nxt(65536); b2_wave = nxt(128);
        W1_trans = nxt(65536); b1_trans = nxt(512); W2_trans = nxt(65536); b2_trans = nxt(128);
        W1_targ = nxt(65536); b1_targ = nxt(512); W2_targ = nxt(65536); b2_targ = nxt(128);
        gF_wave = nxt(128); bF_wave = nxt(128);
        gF_trans = nxt(128); bF_trans = nxt(128);
        gF_targ = nxt(128); bF_targ = nxt(128);
        ei_wt = (const int*)d_in[4];
        ei_tt = (const int*)d_in[5];
    }

    // workspace: Q | K | V | acc | scores | mOrd | z
    float* Q      = (float*)d_ws;
    float* Kb     = Q  + (size_t)N_ * D_;
    float* Vb     = Kb + (size_t)N_ * D_;
    float* acc    = Vb + (size_t)N_ * D_;
    float* scores = acc + (size_t)N_ * D_;
    unsigned* mOrd = (unsigned*)(scores + (size_t)E_ * H_);
    float* z      = (float*)(mOrd + (size_t)N_ * H_);

    float* out_w = (float*)d_out;
    float* out_t = out_w + (size_t)N_ * D_;
    float* out_g = out_t + (size_t)N_ * D_;

    dim3 g128(N_ / 16, D_ / 16), b32(32);
    int initBlocks = (N_ * D_ + 255) / 256;
    int edgeBlocks = (E_ + 7) / 8;     // wave per edge, 8 waves/block
    int rowBlocks  = (N_ + 7) / 8;     // wave per row,  8 waves/block

    // ======== wave -> transition attention ========
    init_mza_kernel<<<initBlocks, 256, 0, stream>>>(mOrd, z, acc);
    wmma_gemm_kernel<<<g128, b32, 0, stream>>>(x_trans, Wq_trans, bq_trans, Q,  D_, D_, 0);
    wmma_gemm_kernel<<<g128, b32, 0, stream>>>(x_wave,  Wk_wave,  bk_wave,  Kb, D_, D_, 0);
    wmma_gemm_kernel<<<g128, b32, 0, stream>>>(x_wave,  Wv_wave,  bv_wave,  Vb, D_, D_, 0);
    edge_scores_kernel<<<edgeBlocks, 256, 0, stream>>>(ei_wt, Q, Kb, x_wave, x_trans,
                                                       We_wt, be_wt, scores, mOrd);
    edge_accum_kernel<<<edgeBlocks, 256, 0, stream>>>(ei_wt, scores, mOrd, z, Vb, acc);
    finalize_attn_kernel<<<initBlocks, 256, 0, stream>>>(acc, z);
    wmma_gemm_kernel<<<g128, b32, 0, stream>>>(acc, Wout_trans, bout_trans, Q, D_, D_, 0);
    row_addln_kernel<<<rowBlocks, 256, 0, stream>>>(x_trans, Q, gA_trans, bA_trans, out_t);

    // ======== transition -> target attention ========
    init_mza_kernel<<<initBlocks, 256, 0, stream>>>(mOrd, z, acc);
    wmma_gemm_kernel<<<g128, b32, 0, stream>>>(x_targ,  Wq_targ,  bq_targ,  Q,  D_, D_, 0);
    wmma_gemm_kernel<<<g128, b32, 0, stream>>>(x_trans, Wk_trans, bk_trans, Kb, D_, D_, 0);
    wmma_gemm_kernel<<<g128, b32, 0, stream>>>(x_trans, Wv_trans, bv_trans, Vb, D_, D_, 0);
    edge_scores_kernel<<<edgeBlocks, 256, 0, stream>>>(ei_tt, Q, Kb, x_trans, x_targ,
                                                       We_tt, be_tt, scores, mOrd);
    edge_accum_kernel<<<edgeBlocks, 256, 0, stream>>>(ei_tt, scores, mOrd, z, Vb, acc);
    finalize_attn_kernel<<<initBlocks, 256, 0, stream>>>(acc, z);
    wmma_gemm_kernel<<<g128, b32, 0, stream>>>(acc, Wout_targ, bout_targ, Q, D_, D_, 0);
    row_addln_kernel<<<rowBlocks, 256, 0, stream>>>(x_targ, Q, gA_targ, bA_targ, out_g);

    // ======== wave: residual-only LN ========
    row_addln_kernel<<<rowBlocks, 256, 0, stream>>>(x_wave, (const float*)nullptr,
                                                    gA_wave, bA_wave, out_w);

    // ======== fused FFN + residual + LN (in place on d_out slices) ========
    ffn_ln_kernel<<<N_ / 16, 128, 0, stream>>>(out_w, W1_wave,  b1_wave,  W2_wave,  b2_wave,  gF_wave,  bF_wave);
    ffn_ln_kernel<<<N_ / 16, 128, 0, stream>>>(out_t, W1_trans, b1_trans, W2_trans, b2_trans, gF_trans, bF_trans);
    ffn_ln_kernel<<<N_ / 16, 128, 0, stream>>>(out_g, W1_targ,  b1_targ,  W2_targ,  b2_targ,  gF_targ,  bF_targ);

    (void)in_sizes; (void)n_in; (void)out_size; (void)ws_size;
}